// StudentModel_33629593927820
// MI455X (gfx1250) — compile-verified
//
#include <hip/hip_runtime.h>
#include <math.h>

typedef float vf2 __attribute__((ext_vector_type(2)));
typedef float vf8 __attribute__((ext_vector_type(8)));

#define NNODES 10000

// ---------------------------------------------------------------------------
// helpers
// ---------------------------------------------------------------------------
__device__ __forceinline__ float atomic_max_f32(float* addr, float val) {
  unsigned int* ua = (unsigned int*)addr;
  unsigned int old = *ua;
  while (__uint_as_float(old) < val) {
    unsigned int assumed = old;
    old = atomicCAS(ua, assumed, __float_as_uint(val));
    if (old == assumed) break;
  }
  return __uint_as_float(old);
}

__device__ __forceinline__ void edge_ends(const long long* __restrict__ ei,
                                          int E, int e, int& s, int& d) {
  if (e < E) {
    s = (int)ei[e];
    d = (int)ei[(size_t)E + e];
  } else {
    s = d = e - E;  // self loops appended after real edges
  }
}

// ---------------------------------------------------------------------------
// zero-padding copy: src[rows x cols] -> dst[drows x dcols], zero elsewhere
// ---------------------------------------------------------------------------
__global__ void pad_matrix(const float* __restrict__ src, float* __restrict__ dst,
                           int rows, int cols, int drows, int dcols) {
  int i = blockIdx.x * blockDim.x + threadIdx.x;
  int total = drows * dcols;
  if (i >= total) return;
  int r = i / dcols, c = i - r * dcols;
  dst[i] = (r < rows && c < cols) ? src[(size_t)r * cols + c] : 0.0f;
}

// ---------------------------------------------------------------------------
// f32 WMMA GEMM, LDS-tiled: D[M x Nn] = A[M x K] * B[K x Nn], row-major.
// Block = 256 threads (8 waves), macro-tile 128 rows x 64 cols, K-chunk 32.
// B chunk staged in LDS K-PAIRED: Bs[(k/2)*TN + col] = {B[k][col],B[k+1][col]}
// so each lane's B fragment (K = k+2*half, k+2*half+1 which is always an even
// pair) is a single aligned ds_load_b64 straight into an even VGPR pair — no
// repacking movs before the WMMA. 8 KB LDS, reused by 8 waves x 4 N-tiles.
// Fragment layouts per ISA 7.12.2:
//   A 16x4: lanes 0-15 hold K=k,k+1; lanes 16-31 hold K=k+2,k+3
//   B 4x16: mirrors A across lanes; C/D: VGPR v -> row v / v+8 by lane half.
// EXEC is all-ones at every WMMA (store guards come after the K loop).
// ---------------------------------------------------------------------------
#define TM 128
#define TN 64
#define TK 32

__global__ void __launch_bounds__(256)
wmma_gemm_f32_tiled(const float* __restrict__ A, const float* __restrict__ B,
                    float* __restrict__ D, int M, int Nn, int K) {
  __shared__ vf2 Bs[(TK / 2) * TN];  // 8 KB, K-paired layout
  const int t    = threadIdx.x;
  const int wave = t >> 5;
  const int lane = t & 31;
  const int half = lane >> 4;
  const int l    = lane & 15;
  const int m0   = blockIdx.y * TM + wave * 16;
  const int nblk = blockIdx.x * TN;

  const float* arow = A + (size_t)(m0 + l) * K + 2 * half;
  float* BsF = (float*)Bs;

  vf8 acc[4] = {};
  for (int k0 = 0; k0 < K; k0 += TK) {
    // stage B[k0:k0+TK, nblk:nblk+TN] into LDS, K-paired
    for (int i = t; i < TK * TN; i += 256) {
      int r = i >> 6, c = i & 63;
      BsF[(((r >> 1) * TN + c) << 1) | (r & 1)] =
          B[(size_t)(k0 + r) * Nn + nblk + c];
    }
    // prefetch next B chunk while this one is consumed (global_prefetch_b8)
    if (k0 + TK < K)
      __builtin_prefetch(&B[(size_t)(k0 + TK + (t >> 6)) * Nn + nblk + (t & 63)], 0, 1);
    __syncthreads();

#pragma unroll
    for (int k = 0; k < TK; k += 4) {
      vf2 a;
      a.x = arow[k0 + k];
      a.y = arow[k0 + k + 1];
      const vf2* brow = Bs + (k / 2 + half) * TN + l;  // K-pair (k+2*half, +1)
#pragma unroll
      for (int nt = 0; nt < 4; ++nt) {
        vf2 b = brow[nt * 16];  // single ds_load_b64
        acc[nt] = __builtin_amdgcn_wmma_f32_16x16x4_f32(false, a, false, b,
                                                        (short)0, acc[nt],
                                                        false, false);
      }
    }
    __syncthreads();
  }

#pragma unroll
  for (int nt = 0; nt < 4; ++nt) {
#pragma unroll
    for (int v = 0; v < 8; ++v) {
      int row = m0 + 8 * half + v;
      if (row < M)
        D[(size_t)row * Nn + nblk + nt * 16 + l] = acc[nt][v];
    }
  }
}

// ---------------------------------------------------------------------------
// attention coefficients: als[n,h] = <H[n,h,:], a_src[h,:]>, same for ald
// ---------------------------------------------------------------------------
__global__ void attn_coeff(const float* __restrict__ H, const float* __restrict__ a_src,
                           const float* __restrict__ a_dst, float* __restrict__ als,
                           float* __restrict__ ald, int heads, int C, int strideH) {
  int i = blockIdx.x * blockDim.x + threadIdx.x;
  if (i >= NNODES * heads) return;
  int n = i / heads, h = i - n * heads;
  const float* hp = H + (size_t)n * strideH + h * C;
  const float* ps = a_src + h * C;
  const float* pd = a_dst + h * C;
  float s = 0.0f, d = 0.0f;
  for (int c = 0; c < C; ++c) {
    float v = hp[c];
    s += v * ps[c];
    d += v * pd[c];
  }
  als[i] = s;
  ald[i] = d;
}

__global__ void fill_f32(float* __restrict__ p, float v, int n) {
  int i = blockIdx.x * blockDim.x + threadIdx.x;
  if (i < n) p[i] = v;
}

// ---------------------------------------------------------------------------
// segment max of leaky_relu(als[src]+ald[dst]) over dst
// ---------------------------------------------------------------------------
__global__ void edge_max(const long long* __restrict__ ei, int E, int Etot,
                         const float* __restrict__ als, const float* __restrict__ ald,
                         float* __restrict__ m, int heads) {
  int i = blockIdx.x * blockDim.x + threadIdx.x;
  if (i >= Etot * heads) return;
  int e = i / heads, h = i - e * heads;
  int s, d;
  edge_ends(ei, E, e, s, d);
  float v = als[s * heads + h] + ald[d * heads + h];
  v = v > 0.0f ? v : 0.2f * v;
  atomic_max_f32(&m[d * heads + h], v);
}

__global__ void fix_m(float* __restrict__ m, int n) {
  int i = blockIdx.x * blockDim.x + threadIdx.x;
  if (i >= n) return;
  float v = m[i];
  if (!(v > -1e30f)) m[i] = 0.0f;  // -inf (no in-edges) -> 0
}

// ---------------------------------------------------------------------------
// e = exp(logit - m[dst]); store per-edge, segment-sum into s[dst]
// ---------------------------------------------------------------------------
__global__ void edge_exp_sum(const long long* __restrict__ ei, int E, int Etot,
                             const float* __restrict__ als, const float* __restrict__ ald,
                             const float* __restrict__ m, float* __restrict__ ssum,
                             float* __restrict__ eexp, int heads) {
  int i = blockIdx.x * blockDim.x + threadIdx.x;
  if (i >= Etot * heads) return;
  int e = i / heads, h = i - e * heads;
  int s, d;
  edge_ends(ei, E, e, s, d);
  float v = als[s * heads + h] + ald[d * heads + h];
  v = v > 0.0f ? v : 0.2f * v;
  float ex = expf(v - m[d * heads + h]);
  eexp[(size_t)e * heads + h] = ex;
  atomicAdd(&ssum[d * heads + h], ex);
}

// ---------------------------------------------------------------------------
// message aggregation: agg[dst, c] += eexp[e, c/C] * H[src, c]  (wave / edge)
// (division by segment-sum deferred to the finalize kernel)
// ---------------------------------------------------------------------------
__global__ void edge_agg(const long long* __restrict__ ei, int E, int Etot,
                         const float* __restrict__ H, const float* __restrict__ eexp,
                         float* __restrict__ agg, int heads, int C, int Ctot,
                         int strideH, int strideA) {
  int wave = (blockIdx.x * blockDim.x + threadIdx.x) >> 5;
  int lane = threadIdx.x & 31;
  if (wave >= Etot) return;
  int s, d;
  edge_ends(ei, E, wave, s, d);
  const float* hs = H + (size_t)s * strideH;
  const float* ee = eexp + (size_t)wave * heads;
  float* ad = agg + (size_t)d * strideA;
  for (int c = lane; c < Ctot; c += 32) {
    int h = c / C;
    atomicAdd(&ad[c], ee[h] * hs[c]);
  }
}

// ---------------------------------------------------------------------------
// fused: y = agg/s + bias + residual; LayerNorm(y)*g+b; ELU -> out  (F == 512)
// ---------------------------------------------------------------------------
__global__ void __launch_bounds__(256)
finalize_ln_elu(const float* __restrict__ agg, const float* __restrict__ ssum,
                const float* __restrict__ bias, const float* __restrict__ resid,
                const float* __restrict__ gamma, const float* __restrict__ beta,
                float* __restrict__ out, int heads, int C) {
  const int F = heads * C;  // 512
  __shared__ float sh[512];
  __shared__ float red[256];
  int n = blockIdx.x, t = threadIdx.x;

  float local = 0.0f;
  for (int c = t; c < F; c += 256) {
    float y = agg[(size_t)n * F + c] / (ssum[n * heads + c / C] + 1e-16f)
            + bias[c] + resid[(size_t)n * F + c];
    sh[c] = y;
    local += y;
  }
  red[t] = local;
  __syncthreads();
  for (int s = 128; s > 0; s >>= 1) {
    if (t < s) red[t] += red[t + s];
    __syncthreads();
  }
  float mu = red[0] / F;
  __syncthreads();

  float lv = 0.0f;
  for (int c = t; c < F; c += 256) {
    float dd = sh[c] - mu;
    lv += dd * dd;
  }
  red[t] = lv;
  __syncthreads();
  for (int s = 128; s > 0; s >>= 1) {
    if (t < s) red[t] += red[t + s];
    __syncthreads();
  }
  float rstd = rsqrtf(red[0] / F + 1e-5f);

  for (int c = t; c < F; c += 256) {
    float y = (sh[c] - mu) * rstd * gamma[c] + beta[c];
    out[(size_t)n * F + c] = y > 0.0f ? y : (expf(y) - 1.0f);
  }
}

// ---------------------------------------------------------------------------
// layer-3 epilogue: out[n,c] = agg[n,c]/s[n] + b3[c] + R3[n,c]   (121 cols)
// ---------------------------------------------------------------------------
__global__ void finalize_out(const float* __restrict__ agg, const float* __restrict__ ssum,
                             const float* __restrict__ bias, const float* __restrict__ resid,
                             float* __restrict__ out) {
  int i = blockIdx.x * blockDim.x + threadIdx.x;
  if (i >= NNODES * 121) return;
  int n = i / 121, c = i - n * 121;
  out[i] = agg[(size_t)n * 128 + c] / (ssum[n] + 1e-16f) + bias[c]
         + resid[(size_t)n * 128 + c];
}

// ---------------------------------------------------------------------------
// host driver
// ---------------------------------------------------------------------------
extern "C" void kernel_launch(void* const* d_in, const int* in_sizes, int n_in,
                              void* d_out, int out_size, void* d_ws, size_t ws_size,
                              hipStream_t stream) {
  (void)n_in; (void)out_size; (void)ws_size;

  const float*     x   = (const float*)d_in[0];
  const long long* ei  = (const long long*)d_in[1];
  const float* W1  = (const float*)d_in[2];
  const float* as1 = (const float*)d_in[3];
  const float* ad1 = (const float*)d_in[4];
  const float* b1  = (const float*)d_in[5];
  const float* rw1 = (const float*)d_in[6];
  const float* g1  = (const float*)d_in[7];
  const float* bt1 = (const float*)d_in[8];
  const float* W2  = (const float*)d_in[9];
  const float* as2 = (const float*)d_in[10];
  const float* ad2 = (const float*)d_in[11];
  const float* b2  = (const float*)d_in[12];
  const float* rw2 = (const float*)d_in[13];
  const float* g2  = (const float*)d_in[14];
  const float* bt2 = (const float*)d_in[15];
  const float* W3  = (const float*)d_in[16];
  const float* as3 = (const float*)d_in[17];
  const float* ad3 = (const float*)d_in[18];
  const float* b3  = (const float*)d_in[19];
  const float* rw3 = (const float*)d_in[20];
  float* out = (float*)d_out;

  const int E    = in_sizes[1] / 2;  // 160000 real edges
  const int Etot = E + NNODES;       // + self loops

  // workspace layout (floats); OOB A-reads in the GEMM stay inside this pool
  float* ws = (float*)d_ws;
  const size_t NB = (size_t)NNODES * 512;
  float* X     = ws;                              // features (in-place per layer)
  float* H     = X + NB;                          // X @ W
  float* R     = H + NB;                          // X @ rW (residual)
  float* AGG   = R + NB;                          // segment-summed messages
  float* ALS   = AGG + NB;                        // [N, heads]
  float* ALD   = ALS + (size_t)NNODES * 8;
  float* MX    = ALD + (size_t)NNODES * 8;
  float* SX    = MX + (size_t)NNODES * 8;
  float* EEXP  = SX + (size_t)NNODES * 8;         // [Etot, heads]
  float* WPAD  = EEXP + (size_t)Etot * 8;         // up to 512*128
  float* RWPAD = WPAD + 65536;

  auto cdiv = [](int a, int b) { return (a + b - 1) / b; };
  const int T = 256;
  const int GY = cdiv(NNODES, TM);  // 79 row-blocks of 128

  // ===================== Layer 1: 50 -> 8 x 64, concat =====================
  pad_matrix<<<cdiv(NNODES * 64, T), T, 0, stream>>>(x, X, NNODES, 50, NNODES, 64);
  pad_matrix<<<cdiv(64 * 512, T), T, 0, stream>>>(W1, WPAD, 50, 512, 64, 512);
  pad_matrix<<<cdiv(64 * 512, T), T, 0, stream>>>(rw1, RWPAD, 50, 512, 64, 512);
  {
    dim3 g(512 / TN, GY);
    wmma_gemm_f32_tiled<<<g, 256, 0, stream>>>(X, WPAD, H, NNODES, 512, 64);
    wmma_gemm_f32_tiled<<<g, 256, 0, stream>>>(X, RWPAD, R, NNODES, 512, 64);
  }
  attn_coeff<<<cdiv(NNODES * 8, T), T, 0, stream>>>(H, as1, ad1, ALS, ALD, 8, 64, 512);
  fill_f32<<<cdiv(NNODES * 8, T), T, 0, stream>>>(MX, -INFINITY, NNODES * 8);
  hipMemsetAsync(SX, 0, (size_t)NNODES * 8 * sizeof(float), stream);
  hipMemsetAsync(AGG, 0, NB * sizeof(float), stream);
  edge_max<<<cdiv(Etot * 8, T), T, 0, stream>>>(ei, E, Etot, ALS, ALD, MX, 8);
  fix_m<<<cdiv(NNODES * 8, T), T, 0, stream>>>(MX, NNODES * 8);
  edge_exp_sum<<<cdiv(Etot * 8, T), T, 0, stream>>>(ei, E, Etot, ALS, ALD, MX, SX, EEXP, 8);
  edge_agg<<<cdiv(Etot * 32, T), T, 0, stream>>>(ei, E, Etot, H, EEXP, AGG, 8, 64, 512, 512, 512);
  finalize_ln_elu<<<NNODES, T, 0, stream>>>(AGG, SX, b1, R, g1, bt1, X, 8, 64);

  // ===================== Layer 2: 512 -> 8 x 64, concat ====================
  {
    dim3 g(512 / TN, GY);
    wmma_gemm_f32_tiled<<<g, 256, 0, stream>>>(X, W2, H, NNODES, 512, 512);
    wmma_gemm_f32_tiled<<<g, 256, 0, stream>>>(X, rw2, R, NNODES, 512, 512);
  }
  attn_coeff<<<cdiv(NNODES * 8, T), T, 0, stream>>>(H, as2, ad2, ALS, ALD, 8, 64, 512);
  fill_f32<<<cdiv(NNODES * 8, T), T, 0, stream>>>(MX, -INFINITY, NNODES * 8);
  hipMemsetAsync(SX, 0, (size_t)NNODES * 8 * sizeof(float), stream);
  hipMemsetAsync(AGG, 0, NB * sizeof(float), stream);
  edge_max<<<cdiv(Etot * 8, T), T, 0, stream>>>(ei, E, Etot, ALS, ALD, MX, 8);
  fix_m<<<cdiv(NNODES * 8, T), T, 0, stream>>>(MX, NNODES * 8);
  edge_exp_sum<<<cdiv(Etot * 8, T), T, 0, stream>>>(ei, E, Etot, ALS, ALD, MX, SX, EEXP, 8);
  edge_agg<<<cdiv(Etot * 32, T), T, 0, stream>>>(ei, E, Etot, H, EEXP, AGG, 8, 64, 512, 512, 512);
  finalize_ln_elu<<<NNODES, T, 0, stream>>>(AGG, SX, b2, R, g2, bt2, X, 8, 64);

  // ===================== Layer 3: 512 -> 121, 1 head =======================
  pad_matrix<<<cdiv(512 * 128, T), T, 0, stream>>>(W3, WPAD, 512, 121, 512, 128);
  pad_matrix<<<cdiv(512 * 128, T), T, 0, stream>>>(rw3, RWPAD, 512, 121, 512, 128);
  {
    dim3 g(128 / TN, GY);
    wmma_gemm_f32_tiled<<<g, 256, 0, stream>>>(X, WPAD, H, NNODES, 128, 512);
    wmma_gemm_f32_tiled<<<g, 256, 0, stream>>>(X, RWPAD, R, NNODES, 128, 512);
  }
  attn_coeff<<<cdiv(NNODES, T), T, 0, stream>>>(H, as3, ad3, ALS, ALD, 1, 121, 128);
  fill_f32<<<cdiv(NNODES, T), T, 0, stream>>>(MX, -INFINITY, NNODES);
  hipMemsetAsync(SX, 0, (size_t)NNODES * sizeof(float), stream);
  hipMemsetAsync(AGG, 0, (size_t)NNODES * 128 * sizeof(float), stream);
  edge_max<<<cdiv(Etot, T), T, 0, stream>>>(ei, E, Etot, ALS, ALD, MX, 1);
  fix_m<<<cdiv(NNODES, T), T, 0, stream>>>(MX, NNODES);
  edge_exp_sum<<<cdiv(Etot, T), T, 0, stream>>>(ei, E, Etot, ALS, ALD, MX, SX, EEXP, 1);
  edge_agg<<<cdiv(Etot * 32, T), T, 0, stream>>>(ei, E, Etot, H, EEXP, AGG, 1, 121, 121, 128, 128);
  finalize_out<<<cdiv(NNODES * 121, T), T, 0, stream>>>(AGG, SX, b3, R, out);
}